// GraphConvolution_11836929868622
// MI455X (gfx1250) — compile-verified
//
#include <hip/hip_runtime.h>
#include <hip/hip_bf16.h>

typedef float v2f __attribute__((ext_vector_type(2)));
typedef float v8f __attribute__((ext_vector_type(8)));

#define KDIM   512
#define NDIM   128
#define KCHUNK 64   // 64x128 f32 = 32 KB LDS per chunk

// ---------------------------------------------------------------------------
// out[n, f] = bias[f]  (fold bias in before accumulation)
// ---------------------------------------------------------------------------
__global__ void gcn_init_bias(const float* __restrict__ bias,
                              float* __restrict__ out, int total4) {
    int i = blockIdx.x * blockDim.x + threadIdx.x;   // one float4 per thread
    if (i < total4) {
        int f = (i * 4) & (NDIM - 1);
        float4 b = *(const float4*)(bias + f);
        *(float4*)(out + (size_t)i * 4) = b;
    }
}

// ---------------------------------------------------------------------------
// P = X @ B   using V_WMMA_F32_16X16X4_F32 (exact fp32 path).
// One wave computes a 16-row x 128-col strip; B staged via 32 KB LDS chunks.
// ---------------------------------------------------------------------------
__global__ __launch_bounds__(256)
void gcn_gemm_wmma(const float* __restrict__ X, const float* __restrict__ B,
                   float* __restrict__ P, int nRowBlocks) {
    __shared__ float Bs[KCHUNK * NDIM];

    const int tid  = threadIdx.x;
    const int lane = tid & 31;
    const int wave = tid >> 5;        // 0..7
    const int half = lane >> 4;       // 0/1 -> K pair select
    const int l16  = lane & 15;       // M (A) / N (B,C)

    const int  rb     = blockIdx.x * 8 + wave;     // wave-uniform
    const bool active = (rb < nRowBlocks);
    const size_t row0 = (size_t)rb * 16;

    v8f zero = {};
    v8f acc[8];
#pragma unroll
    for (int t = 0; t < 8; ++t) acc[t] = zero;

    const float* xrow = active ? (X + (row0 + l16) * KDIM + half * 2) : X;

    for (int kc = 0; kc < KDIM; kc += KCHUNK) {
        __syncthreads();
        {
            const float4* src = (const float4*)(B + (size_t)kc * NDIM);
            float4*       dst = (float4*)Bs;
            for (int i = tid; i < (KCHUNK * NDIM) / 4; i += 256) dst[i] = src[i];
        }
        __syncthreads();

        if (active) {
#pragma unroll 4
            for (int k = 0; k < KCHUNK; k += 4) {
                v2f a = *(const v2f*)(xrow + kc + k);   // global_load_b64
#pragma unroll
                for (int t = 0; t < 8; ++t) {
                    int base = (k + half * 2) * NDIM + t * 16 + l16;
                    v2f b;
                    b.x = Bs[base];          // fused -> ds_load_2addr_b32
                    b.y = Bs[base + NDIM];
                    acc[t] = __builtin_amdgcn_wmma_f32_16x16x4_f32(
                        false, a, false, b, (short)0, acc[t], false, false);
                }
            }
        }
    }

    if (active) {
#pragma unroll
        for (int t = 0; t < 8; ++t) {
#pragma unroll
            for (int v = 0; v < 8; ++v) {
                P[(row0 + v + 8 * half) * NDIM + t * 16 + l16] = acc[t][v];
            }
        }
    }
}

// ---------------------------------------------------------------------------
// CSR build step 1: histogram of destination rows (int atomics, cheap)
// ---------------------------------------------------------------------------
__global__ void gcn_count_rows(const int* __restrict__ rows,
                               int* __restrict__ cnt, int nEdges) {
    int e = blockIdx.x * blockDim.x + threadIdx.x;
    if (e < nEdges) atomicAdd(&cnt[rows[e]], 1);
}

// ---------------------------------------------------------------------------
// CSR build step 2: exclusive scan over nRows bins (single 1024-thread block).
// Writes rowPtr[0..n] and a working cursor copy cur[0..n-1].
// ---------------------------------------------------------------------------
__global__ __launch_bounds__(1024)
void gcn_scan_rows(const int* __restrict__ cnt, int* __restrict__ rowPtr,
                   int* __restrict__ cur, int n) {
    __shared__ int partial[1024];
    const int tid   = threadIdx.x;
    const int chunk = (n + 1023) / 1024;
    const int start = tid * chunk;
    const int end   = (start + chunk < n) ? (start + chunk) : n;

    int sum = 0;
    for (int i = start; i < end; ++i) sum += cnt[i];
    partial[tid] = sum;
    __syncthreads();

    // Hillis-Steele inclusive scan over the 1024 partials
    for (int off = 1; off < 1024; off <<= 1) {
        int v     = partial[tid];
        int vprev = (tid >= off) ? partial[tid - off] : 0;
        __syncthreads();
        partial[tid] = v + vprev;
        __syncthreads();
    }
    int run = (tid == 0) ? 0 : partial[tid - 1];   // exclusive base for chunk

    for (int i = start; i < end; ++i) {
        rowPtr[i] = run;
        cur[i]    = run;
        run += cnt[i];
    }
    if (start < n && end == n) rowPtr[n] = run;
}

// ---------------------------------------------------------------------------
// CSR build step 3: cursor scatter of (val, col) into row-sorted arrays
// ---------------------------------------------------------------------------
__global__ void gcn_scatter_edges(const float* __restrict__ vals,
                                  const int*   __restrict__ rows,
                                  const int*   __restrict__ cols,
                                  int* __restrict__ cur,
                                  float* __restrict__ sVal,
                                  int*   __restrict__ sCol, int nEdges) {
    int e = blockIdx.x * blockDim.x + threadIdx.x;
    if (e < nEdges) {
        int pos   = atomicAdd(&cur[rows[e]], 1);
        sVal[pos] = vals[e];
        sCol[pos] = cols[e];
    }
}

// ---------------------------------------------------------------------------
// Gather SpMM: one wave owns one output row; lanes hold 4 features each.
// out[r, :] += sum_{e in row r} sVal[e] * P[sCol[e], :]   (no fp32 atomics)
// ---------------------------------------------------------------------------
__global__ void gcn_spmm_gather(const int*   __restrict__ rowPtr,
                                const float* __restrict__ sVal,
                                const int*   __restrict__ sCol,
                                const float* __restrict__ P,
                                float* __restrict__ out, int nRows) {
    const int lane   = threadIdx.x & 31;
    const int gwave  = (blockIdx.x * blockDim.x + threadIdx.x) >> 5;
    const int nWaves = (gridDim.x * blockDim.x) >> 5;

    for (int r = gwave; r < nRows; r += nWaves) {
        const int beg = rowPtr[r];
        const int end = rowPtr[r + 1];
        float4 acc = make_float4(0.f, 0.f, 0.f, 0.f);
        for (int i = beg; i < end; ++i) {
            float  v = sVal[i];                       // wave-broadcast
            size_t c = (size_t)sCol[i];
            float4 p = *(const float4*)(P + c * NDIM + lane * 4);
            acc.x += v * p.x; acc.y += v * p.y;
            acc.z += v * p.z; acc.w += v * p.w;
        }
        float* o = out + (size_t)r * NDIM + lane * 4;
        float4 cv = *(const float4*)o;
        cv.x += acc.x; cv.y += acc.y; cv.z += acc.z; cv.w += acc.w;
        *(float4*)o = cv;                              // exclusive row owner
    }
}

// ---------------------------------------------------------------------------
__global__ void gcn_relu(float* __restrict__ out, int total4) {
    int i = blockIdx.x * blockDim.x + threadIdx.x;
    if (i < total4) {
        float4* p = (float4*)(out + (size_t)i * 4);
        float4  v = *p;
        v.x = fmaxf(v.x, 0.f); v.y = fmaxf(v.y, 0.f);
        v.z = fmaxf(v.z, 0.f); v.w = fmaxf(v.w, 0.f);
        *p = v;
    }
}

// ---------------------------------------------------------------------------
extern "C" void kernel_launch(void* const* d_in, const int* in_sizes, int n_in,
                              void* d_out, int out_size, void* d_ws, size_t ws_size,
                              hipStream_t stream) {
    const float* x       = (const float*)d_in[0];   // [N, 512]
    const float* kernels = (const float*)d_in[1];   // [2, 512, 128]
    const float* bias    = (const float*)d_in[2];   // [128]
    const float* avals   = (const float*)d_in[3];   // [2, E]
    const int*   arows   = (const int*)  d_in[4];   // [2, E]
    const int*   acols   = (const int*)  d_in[5];   // [2, E]
    float*       out     = (float*)d_out;           // [N, 128]

    const int nNodes    = in_sizes[0] / KDIM;          // 50000
    const int nSupports = in_sizes[1] / (KDIM * NDIM); // 2
    const int nEdges    = in_sizes[3] / nSupports;     // 800000
    const int total4    = (nNodes * NDIM) / 4;
    const int rowBlocks = nNodes / 16;

    // -------- workspace layout (reused serially across supports) --------
    char* ws = (char*)d_ws;
    float* pre    = (float*)ws;                 ws += (size_t)nNodes * NDIM * 4;
    int*   cnt    = (int*)ws;                   ws += (size_t)nNodes * 4;
    int*   rowPtr = (int*)ws;                   ws += (size_t)(nNodes + 1) * 4;
    int*   cur    = (int*)ws;                   ws += (size_t)nNodes * 4;
    float* sVal   = (float*)ws;                 ws += (size_t)nEdges * 4;
    int*   sCol   = (int*)ws;

    gcn_init_bias<<<(total4 + 255) / 256, 256, 0, stream>>>(bias, out, total4);

    const int gemmBlocks = (rowBlocks + 7) / 8;
    const int edgeBlocks = (nEdges + 255) / 256;
    const int rowWaveBlk = (nNodes * 32 + 255) / 256;   // one wave per row

    for (int s = 0; s < nSupports; ++s) {
        const float* vals_s = avals + (size_t)s * nEdges;
        const int*   rows_s = arows + (size_t)s * nEdges;
        const int*   cols_s = acols + (size_t)s * nEdges;

        // dense GEMM (fp32 WMMA): pre = x @ kernels[s]
        gcn_gemm_wmma<<<gemmBlocks, 256, 0, stream>>>(
            x, kernels + (size_t)s * KDIM * NDIM, pre, rowBlocks);

        // COO -> CSR (int atomics only)
        hipMemsetAsync(cnt, 0, (size_t)nNodes * 4, stream);
        gcn_count_rows<<<edgeBlocks, 256, 0, stream>>>(rows_s, cnt, nEdges);
        gcn_scan_rows<<<1, 1024, 0, stream>>>(cnt, rowPtr, cur, nNodes);
        gcn_scatter_edges<<<edgeBlocks, 256, 0, stream>>>(
            vals_s, rows_s, cols_s, cur, sVal, sCol, nEdges);

        // gather SpMM, accumulate into out (race-free, no float atomics)
        gcn_spmm_gather<<<rowWaveBlk, 256, 0, stream>>>(
            rowPtr, sVal, sCol, pre, out, nNodes);
    }

    gcn_relu<<<(total4 + 255) / 256, 256, 0, stream>>>(out, total4);
}